// DeformConv_30090540875951
// MI455X (gfx1250) — compile-verified
//
#include <hip/hip_runtime.h>
#include <math.h>

typedef __attribute__((ext_vector_type(2))) float v2f;
typedef __attribute__((ext_vector_type(8))) float v8f;

#define BATCH 4
#define CINCH 128
#define HH 128
#define WWW 128
#define NPIX (HH * WWW)          // 16384
#define COUTCH 128
#define OFFCH 54                 // 18 y-off + 18 x-off + 18 mask
#define OFFPAD 64                // padded M for WMMA
#define KTOT (CINCH * 9)         // 1152
#define CPC 8                    // channels per K-chunk
#define KC (CPC * 9)             // 72 K values per chunk
#define NCHUNK (CINCH / CPC)     // 16
#define NT 32                    // pixel tile per workgroup

struct Meta { float w0, w1, w2, w3; int i0, i1, i2, i3; };

// ---------------------------------------------------------------------------
// Kernel 0: zero-pad offset weights 54x1152 -> 64x1152 (avoids OOB A reads).
// ---------------------------------------------------------------------------
__global__ __launch_bounds__(256) void pad_woff_kernel(
    const float* __restrict__ w_off, float* __restrict__ wpad)
{
    const int i = blockIdx.x * 256 + threadIdx.x;   // 0 .. 64*1152-1
    if (i >= OFFPAD * KTOT) return;
    const int m = i / KTOT;
    wpad[i] = (m < OFFCH) ? w_off[i] : 0.0f;
}

// ---------------------------------------------------------------------------
// Kernel 1: offset/mask conv as fp32 WMMA GEMM.
//   A = wpad (64 x 1152), B = im2col(x) staged in LDS (72 x 32 per chunk).
//   256 threads = 8 waves = 4 M-tiles x 2 N-tiles, 1 accumulator each.
//   Epilogue: +b_off, sigmoid on mask channels (36..53), store 54 rows.
// ---------------------------------------------------------------------------
__global__ __launch_bounds__(256) void offset_conv_wmma_kernel(
    const float* __restrict__ x, const float* __restrict__ wpad,
    const float* __restrict__ b_off, float* __restrict__ offout)
{
    __shared__ float s_cols[KC * NT];      // 72 x 32 -> 9216 B

    const int tid  = threadIdx.x;
    const int lane = tid & 31;
    const int wave = tid >> 5;
    const int mt   = wave & 3;             // M tile 0..3  (m0 = 16*mt)
    const int nt   = wave >> 2;            // N sub-tile 0..1
    const int m0   = mt * 16;
    const int nsub = nt * 16;
    const int n0   = blockIdx.x * NT;
    const int b    = blockIdx.y;

    const float* xb   = x + (size_t)b * CINCH * NPIX;
    const float* wrow = wpad + (size_t)(m0 + (lane & 15)) * KTOT;
    const int ksel    = (lane >> 4) << 1;  // lanes 16..31 hold K=2,3

    v8f acc = {};

    for (int chunk = 0; chunk < NCHUNK; ++chunk) {
        const int c0 = chunk * CPC;

        // ---- stage im2col tile 72 x 32 (zero-padded borders)
        for (int v = tid; v < KC * NT; v += 256) {
            const int p  = v & (NT - 1);
            const int kl = v >> 5;           // 0..71
            const int cl = kl / 9;
            const int k2 = kl - cl * 9;
            const int ky = k2 / 3;
            const int kx = k2 - ky * 3;
            const int n  = n0 + p;
            const int ho = n >> 7;
            const int wo = n & 127;
            const int y  = ho - 1 + ky;
            const int xx = wo - 1 + kx;
            float val = 0.0f;
            if ((unsigned)y < (unsigned)HH && (unsigned)xx < (unsigned)WWW)
                val = xb[(size_t)(c0 + cl) * NPIX + y * WWW + xx];
            s_cols[kl * NT + p] = val;
        }
        __syncthreads();

        const int kg0 = c0 * 9;
#pragma unroll
        for (int t = 0; t < KC / 4; ++t) {
            v2f a, bb;
            const int kg = kg0 + t * 4 + ksel;
            a.x = wrow[kg];
            a.y = wrow[kg + 1];
            const int kl = t * 4 + ksel;
            const int nn = nsub + (lane & 15);
            bb.x = s_cols[kl * NT + nn];
            bb.y = s_cols[(kl + 1) * NT + nn];
            acc = __builtin_amdgcn_wmma_f32_16x16x4_f32(
                false, a, false, bb, (short)0, acc, false, false);
        }
        __syncthreads();
    }

    // ---- epilogue
    const int nn    = n0 + nsub + (lane & 15);
    const int mbase = m0 + ((lane >> 4) << 3);
#pragma unroll
    for (int r = 0; r < 8; ++r) {
        const int oc = mbase + r;
        if (oc < OFFCH) {
            float v = acc[r] + b_off[oc];
            if (oc >= 36) v = 1.0f / (1.0f + __expf(-v));   // mask sigmoid
            offout[((size_t)(b * OFFCH + oc)) * NPIX + nn] = v;
        }
    }
}

// ---------------------------------------------------------------------------
// Kernel 2: fused deformable-sampling + fp32 WMMA GEMM.
// 256 threads = 8 waves; wave w owns cout rows [16w,16w+16) and BOTH 16-pixel
// sub-tiles (2 accumulators -> each A fragment feeds 2 WMMAs).
// ---------------------------------------------------------------------------
__global__ __launch_bounds__(256) void deform_wmma_kernel(
    const float* __restrict__ x, const float* __restrict__ offout,
    const float* __restrict__ weight, const float* __restrict__ bias,
    float* __restrict__ out)
{
    __shared__ Meta  s_meta[18 * NT];      // 18 combos x 32 px -> 18432 B
    __shared__ float s_cols[KC * NT];      // 72 x 32           ->  9216 B

    const int tid  = threadIdx.x;
    const int lane = tid & 31;
    const int wave = tid >> 5;
    const int n0   = blockIdx.x * NT;
    const int b    = blockIdx.y;
    const int m0   = wave * 16;

    // ---- stage bilinear metadata: fold mask + boundary validity into weights
    for (int e = tid; e < 18 * NT; e += 256) {
        const int p     = e & (NT - 1);
        const int combo = e >> 5;          // og*9 + k2
        const int og    = combo / 9;
        const int k2    = combo - og * 9;
        const int ky    = k2 / 3;
        const int kx    = k2 - ky * 3;
        const int n     = n0 + p;
        const int ho    = n >> 7;
        const int wo    = n & 127;

        const float* ob = offout + (size_t)b * OFFCH * NPIX;
        const float oy = ob[(og * 18 + k2 * 2    ) * NPIX + n];
        const float ox = ob[(og * 18 + k2 * 2 + 1) * NPIX + n];
        const float mk = ob[(36 + og * 9 + k2    ) * NPIX + n];

        const float sy = (float)(ho - 1 + ky) + oy;
        const float sx = (float)(wo - 1 + kx) + ox;
        const float y0f = floorf(sy), x0f = floorf(sx);
        const float ly = sy - y0f, lx = sx - x0f;
        const int y0 = (int)y0f, x0 = (int)x0f;
        const int y1 = y0 + 1,   x1 = x0 + 1;

        const float vy0 = ((unsigned)y0 < (unsigned)HH)  ? 1.f : 0.f;
        const float vy1 = ((unsigned)y1 < (unsigned)HH)  ? 1.f : 0.f;
        const float vx0 = ((unsigned)x0 < (unsigned)WWW) ? 1.f : 0.f;
        const float vx1 = ((unsigned)x1 < (unsigned)WWW) ? 1.f : 0.f;

        const int cy0 = y0 < 0 ? 0 : (y0 > HH - 1  ? HH - 1  : y0);
        const int cy1 = y1 < 0 ? 0 : (y1 > HH - 1  ? HH - 1  : y1);
        const int cx0 = x0 < 0 ? 0 : (x0 > WWW - 1 ? WWW - 1 : x0);
        const int cx1 = x1 < 0 ? 0 : (x1 > WWW - 1 ? WWW - 1 : x1);

        Meta m;
        m.w0 = (1.f - ly) * (1.f - lx) * vy0 * vx0 * mk;
        m.w1 = (1.f - ly) * lx         * vy0 * vx1 * mk;
        m.w2 = ly * (1.f - lx)         * vy1 * vx0 * mk;
        m.w3 = ly * lx                 * vy1 * vx1 * mk;
        m.i0 = cy0 * WWW + cx0;  m.i1 = cy0 * WWW + cx1;
        m.i2 = cy1 * WWW + cx0;  m.i3 = cy1 * WWW + cx1;
        s_meta[e] = m;
    }
    __syncthreads();

    v8f acc0 = {};
    v8f acc1 = {};
    const float* xb   = x + (size_t)b * CINCH * NPIX;
    const float* wrow = weight + (size_t)(m0 + (lane & 15)) * KTOT;
    const int ksel    = (lane >> 4) << 1;

    for (int chunk = 0; chunk < NCHUNK; ++chunk) {
        const int c0 = chunk * CPC;

        // ---- stage 72 x 32 cols tile via bilinear gather
        for (int v = tid; v < KC * NT; v += 256) {
            const int p  = v & (NT - 1);
            const int kl = v >> 5;           // 0..71
            const int cl = kl / 9;
            const int k2 = kl - cl * 9;
            const int c  = c0 + cl;
            const Meta m = s_meta[((c >> 6) * 9 + k2) * NT + p];
            const float* xc = xb + (size_t)c * NPIX;
            s_cols[kl * NT + p] = m.w0 * xc[m.i0] + m.w1 * xc[m.i1] +
                                  m.w2 * xc[m.i2] + m.w3 * xc[m.i3];
        }
        __syncthreads();

        // ---- consume: each A fragment feeds both N sub-tiles
        const int kg0 = c0 * 9;
#pragma unroll
        for (int t = 0; t < KC / 4; ++t) {
            v2f a, b0, b1;
            const int kg = kg0 + t * 4 + ksel;
            a.x = wrow[kg];
            a.y = wrow[kg + 1];
            const int kl = t * 4 + ksel;
            const int nn = lane & 15;
            b0.x = s_cols[kl * NT + nn];
            b0.y = s_cols[(kl + 1) * NT + nn];
            b1.x = s_cols[kl * NT + 16 + nn];
            b1.y = s_cols[(kl + 1) * NT + 16 + nn];
            acc0 = __builtin_amdgcn_wmma_f32_16x16x4_f32(
                false, a, false, b0, (short)0, acc0, false, false);
            acc1 = __builtin_amdgcn_wmma_f32_16x16x4_f32(
                false, a, false, b1, (short)0, acc1, false, false);
        }
        __syncthreads();
    }

    // ---- epilogue: VGPR r -> M=r (lanes 0-15), M=8+r (lanes 16-31)
    const int nn0   = n0 + (lane & 15);
    const int mbase = m0 + ((lane >> 4) << 3);
#pragma unroll
    for (int r = 0; r < 8; ++r) {
        const int mm = mbase + r;
        const float bv = bias[mm];
        float* orow = out + ((size_t)(b * COUTCH + mm)) * NPIX;
        orow[nn0]      = acc0[r] + bv;
        orow[nn0 + 16] = acc1[r] + bv;
    }
}

// ---------------------------------------------------------------------------
extern "C" void kernel_launch(void* const* d_in, const int* in_sizes, int n_in,
                              void* d_out, int out_size, void* d_ws, size_t ws_size,
                              hipStream_t stream) {
    const float* x      = (const float*)d_in[0];
    const float* w_off  = (const float*)d_in[1];
    const float* b_off  = (const float*)d_in[2];
    const float* weight = (const float*)d_in[3];
    const float* bias   = (const float*)d_in[4];
    float* out = (float*)d_out;

    float* offout = (float*)d_ws;                        // B*54*NPIX floats
    float* wpad   = offout + (size_t)BATCH * OFFCH * NPIX; // 64*1152 floats

    pad_woff_kernel<<<(OFFPAD * KTOT + 255) / 256, 256, 0, stream>>>(w_off, wpad);

    dim3 g1(NPIX / NT, BATCH);
    offset_conv_wmma_kernel<<<g1, 256, 0, stream>>>(x, wpad, b_off, offout);

    dim3 g2(NPIX / NT, BATCH);
    deform_wmma_kernel<<<g2, 256, 0, stream>>>(x, offout, weight, bias, out);
}